// SpikingStateSpaceTopologyTransformer_63995012711050
// MI455X (gfx1250) — compile-verified
//
#include <hip/hip_runtime.h>

typedef float f4 __attribute__((ext_vector_type(4)));

namespace {

constexpr int T_STEPS = 128;
constexpr int B = 16, H = 8, V = 25, D = 64;
constexpr int NELEM = B * H * V * D;          // 204800 state elements per timestep
constexpr int NSTAGE = 16;                    // LDS ring depth (timesteps in flight), power of 2
constexpr int BLOCK_THREADS = 128;            // 4 waves (wave32)
constexpr int WAVES_PER_BLOCK = BLOCK_THREADS / 32;
constexpr int GRID_BLOCKS = NELEM / (BLOCK_THREADS * 4);   // float4 per thread -> 400 blocks

template <int K>
__device__ __forceinline__ void wait_async() {
  // ASYNCcnt decrements in order for async loads, so waiting <= K guarantees the
  // oldest (outstanding - K) async-load instructions have landed in LDS.
  asm volatile("s_wait_asynccnt %0" ::"n"(K) : "memory");
}

__device__ __forceinline__ void wait_ds0() {
  // Ensure our ds_load_b128 reads of a ring slot completed before the slot is
  // re-armed with a new async load (async LDS writes are unordered vs DS ops).
  asm volatile("s_wait_dscnt 0" ::: "memory");
}

__device__ __forceinline__ void async_copy16_to_lds(const f4* lds_dst,
                                                    const float* gbase,
                                                    unsigned gl_byte_off) {
  // GVS mode: global address is the 64-bit saddr plus the 32-bit vaddr offset; each lane copies 16B to the LDS address in vdst.
  // q/kv are streamed exactly once (315 MB > 192 MB L2) -> non-temporal.
  unsigned lds_addr = (unsigned)(uintptr_t)lds_dst;  // 32-bit LDS byte offset (addrspace 3)
  asm volatile("global_load_async_to_lds_b128 %0, %1, %2 th:TH_LOAD_NT"
               ::"v"(lds_addr), "v"(gl_byte_off), "s"(gbase)
               : "memory");
}

// Tail drain: t = T-1-R, with R = stages still outstanding after this one.
template <int R, typename F>
__device__ __forceinline__ void drain(F&& f) {
  wait_async<2 * R>();
  f(T_STEPS - 1 - R);
  if constexpr (R > 0) drain<R - 1>(static_cast<F&&>(f));
}

}  // namespace

__global__ __launch_bounds__(BLOCK_THREADS) void retnet_scan_kernel(
    const float* __restrict__ q, const float* __restrict__ kv,
    const float* __restrict__ decay, float* __restrict__ out) {
  // Per-wave private ring: NSTAGE stages x (32 f4 of q + 32 f4 of kv) = 1 KB/stage.
  // 4 waves * 16 stages * 1 KB = 64 KB per block; no barriers needed anywhere.
  __shared__ f4 lds4[WAVES_PER_BLOCK * NSTAGE * 64];

  const int tid  = threadIdx.x;
  const int wave = tid >> 5;
  const int lane = tid & 31;
  const int gtid = blockIdx.x * BLOCK_THREADS + tid;  // float4 index
  const int e0   = gtid * 4;                          // element index within a timestep

  // decay layout (1,1,H,1,D): index = h*D + d ; broadcast over B and V.
  const int h = (e0 / (V * D)) % H;
  f4 dec = *(const f4*)(decay + h * D + (e0 & (D - 1)));
  f4 omd = 1.0f - dec;

  const unsigned elem_byte = (unsigned)e0 * 4u;
  f4* wave_ring = &lds4[wave * NSTAGE * 64];

  auto issue = [&](int t) {
    const int s = t & (NSTAGE - 1);
    f4* slot = wave_ring + s * 64;
    const unsigned goff = (unsigned)t * (unsigned)(NELEM * 4) + elem_byte;
    async_copy16_to_lds(slot + lane,      q,  goff);   // q  -> slot[0..31]
    async_copy16_to_lds(slot + 32 + lane, kv, goff);   // kv -> slot[32..63]
  };

  f4 S = {0.f, 0.f, 0.f, 0.f};

  // Prefill the ring: 2*NSTAGE = 32 async-load instructions outstanding
  // (~16 KB/wave in flight; ~26 MB device-wide across 1600 waves).
  #pragma unroll
  for (int t = 0; t < NSTAGE; ++t) issue(t);

  auto body = [&](int t, bool reissue) {
    const int s = t & (NSTAGE - 1);
    const f4* slot = wave_ring + s * 64;
    f4 qv  = slot[lane];        // ds_load_b128
    f4 kvv = slot[32 + lane];   // ds_load_b128
    wait_ds0();                 // slot drained -> safe to re-arm
    if (reissue) issue(t + NSTAGE);
    S = dec * S + omd * kvv;    // v_fma_f32 x4 (+ v_mul for omd*kv)
    f4 o = qv * S;
    __builtin_nontemporal_store(o, (f4*)(out + (size_t)t * NELEM + e0));
  };

  // Steady state: 2*NSTAGE outstanding; wait <= 2*(NSTAGE-1) => oldest stage landed.
  for (int t = 0; t < T_STEPS - NSTAGE; ++t) {
    wait_async<2 * (NSTAGE - 1)>();
    body(t, true);
  }
  // Drain tail: outstanding shrinks by 2 each step -> tighten the wait 30 -> 0.
  drain<NSTAGE - 1>([&](int t) { body(t, false); });
}

extern "C" void kernel_launch(void* const* d_in, const int* in_sizes, int n_in,
                              void* d_out, int out_size, void* d_ws, size_t ws_size,
                              hipStream_t stream) {
  const float* q     = (const float*)d_in[0];
  const float* kv    = (const float*)d_in[1];
  const float* decay = (const float*)d_in[2];
  float* out = (float*)d_out;

  dim3 grid(GRID_BLOCKS);     // 400
  dim3 block(BLOCK_THREADS);  // 128 = 4 wave32
  hipLaunchKernelGGL(retnet_scan_kernel, grid, block, 0, stream, q, kv, decay, out);
}